// GCNIIConv_72645076845142
// MI455X (gfx1250) — compile-verified
//
#include <hip/hip_runtime.h>
#include <hip/hip_bf16.h>

#define DFEAT 64

typedef __attribute__((ext_vector_type(2))) float v2f;
typedef __attribute__((ext_vector_type(8))) float v8f;

// ---------------------------------------------------------------------------
// Kernel 1: support = 0.9*x + 0.1*h0   (float4 vectorized -> global_load_b128)
// ---------------------------------------------------------------------------
__global__ void gcnii_support_kernel(const float* __restrict__ x,
                                     const float* __restrict__ h0,
                                     float* __restrict__ sup,
                                     int total4) {
    int i = blockIdx.x * blockDim.x + threadIdx.x;
    if (i >= total4) return;
    float4 xv = ((const float4*)x)[i];
    float4 hv = ((const float4*)h0)[i];
    float4 s;
    s.x = 0.9f * xv.x + 0.1f * hv.x;
    s.y = 0.9f * xv.y + 0.1f * hv.y;
    s.z = 0.9f * xv.z + 0.1f * hv.z;
    s.w = 0.9f * xv.w + 0.1f * hv.w;
    ((float4*)sup)[i] = s;
}

// ---------------------------------------------------------------------------
// Kernel 2: out = 0.5 * (support @ weight) via V_WMMA_F32_16X16X4_F32
// Block = 256 threads = 8 waves; each wave owns a 16-row x 64-col output tile.
// Weight (64x64 f32 = 16 KB) staged in LDS once per block.
// ---------------------------------------------------------------------------
__global__ void gcnii_gemm_kernel(const float* __restrict__ sup,
                                  const float* __restrict__ W,
                                  float* __restrict__ out,
                                  int n) {
    __shared__ float Wlds[DFEAT * DFEAT];

    // Cooperative LDS fill: 4096 floats = 1024 float4, 256 threads -> 4 each
    {
        const float4* src = (const float4*)W;
        float4* dst = (float4*)Wlds;
        for (int i = threadIdx.x; i < (DFEAT * DFEAT / 4); i += 256)
            dst[i] = src[i];
    }
    __syncthreads();

    const int wave = threadIdx.x >> 5;   // wave32
    const int lane = threadIdx.x & 31;
    const int rowBase = blockIdx.x * 128 + wave * 16;
    if (rowBase >= n) return;            // wave-uniform guard: EXEC stays all-1s

    const int m     = lane & 15;         // row within A tile / col within B tile
    const int khalf = (lane >> 4) * 2;   // 0 or 2: K sub-pair held by this lane

    const float* arow = sup + (size_t)(rowBase + m) * DFEAT;

    v8f acc0 = {}, acc1 = {}, acc2 = {}, acc3 = {};

#pragma unroll
    for (int kt = 0; kt < 16; ++kt) {    // K = 64 in steps of 4
        const int k0 = kt * 4 + khalf;
        v2f a;
        a.x = arow[k0];
        a.y = arow[k0 + 1];

        v2f b0, b1, b2, b3;
        b0.x = Wlds[k0 * DFEAT +  0 + m];  b0.y = Wlds[(k0 + 1) * DFEAT +  0 + m];
        b1.x = Wlds[k0 * DFEAT + 16 + m];  b1.y = Wlds[(k0 + 1) * DFEAT + 16 + m];
        b2.x = Wlds[k0 * DFEAT + 32 + m];  b2.y = Wlds[(k0 + 1) * DFEAT + 32 + m];
        b3.x = Wlds[k0 * DFEAT + 48 + m];  b3.y = Wlds[(k0 + 1) * DFEAT + 48 + m];

        acc0 = __builtin_amdgcn_wmma_f32_16x16x4_f32(false, a, false, b0, (short)0, acc0, false, false);
        acc1 = __builtin_amdgcn_wmma_f32_16x16x4_f32(false, a, false, b1, (short)0, acc1, false, false);
        acc2 = __builtin_amdgcn_wmma_f32_16x16x4_f32(false, a, false, b2, (short)0, acc2, false, false);
        acc3 = __builtin_amdgcn_wmma_f32_16x16x4_f32(false, a, false, b3, (short)0, acc3, false, false);
    }

    // C/D layout: VGPR v, lanes 0-15 -> M=v ; lanes 16-31 -> M=v+8 ; N = lane&15
    const int rOff = (lane >> 4) * 8;
#pragma unroll
    for (int v = 0; v < 8; ++v) {
        float* orow = out + (size_t)(rowBase + rOff + v) * DFEAT + m;
        orow[ 0] = 0.5f * acc0[v];
        orow[16] = 0.5f * acc1[v];
        orow[32] = 0.5f * acc2[v];
        orow[48] = 0.5f * acc3[v];
    }
}

// ---------------------------------------------------------------------------
// Kernel 3: out[row] += 0.5 * val * support[col]   (scatter-add over edges)
// 16 lanes per edge, float4 per lane -> one coalesced 256B row gather/edge.
// ---------------------------------------------------------------------------
__global__ void gcnii_edge_kernel(const int* __restrict__ erow,
                                  const int* __restrict__ ecol,
                                  const float* __restrict__ eval,
                                  const float* __restrict__ sup,
                                  float* __restrict__ out,
                                  int numE) {
    int gid = blockIdx.x * blockDim.x + threadIdx.x;
    int e      = gid >> 4;
    int lane16 = gid & 15;
    if (e >= numE) return;

    int   r = erow[e];
    int   c = ecol[e];
    float v = 0.5f * eval[e];

    float4 s = ((const float4*)(sup + (size_t)c * DFEAT))[lane16];
    float* dst = out + (size_t)r * DFEAT + lane16 * 4;

    __hip_atomic_fetch_add(dst + 0, v * s.x, __ATOMIC_RELAXED, __HIP_MEMORY_SCOPE_AGENT);
    __hip_atomic_fetch_add(dst + 1, v * s.y, __ATOMIC_RELAXED, __HIP_MEMORY_SCOPE_AGENT);
    __hip_atomic_fetch_add(dst + 2, v * s.z, __ATOMIC_RELAXED, __HIP_MEMORY_SCOPE_AGENT);
    __hip_atomic_fetch_add(dst + 3, v * s.w, __ATOMIC_RELAXED, __HIP_MEMORY_SCOPE_AGENT);
}

// ---------------------------------------------------------------------------
extern "C" void kernel_launch(void* const* d_in, const int* in_sizes, int n_in,
                              void* d_out, int out_size, void* d_ws, size_t ws_size,
                              hipStream_t stream) {
    const float* x    = (const float*)d_in[0];
    const float* h0   = (const float*)d_in[1];
    const int*   erow = (const int*)  d_in[2];
    const int*   ecol = (const int*)  d_in[3];
    const float* eval = (const float*)d_in[4];
    const float* W    = (const float*)d_in[5];

    const int N = in_sizes[0] / DFEAT;
    const int E = in_sizes[2];

    float* sup = (float*)d_ws;          // N*64 floats scratch
    float* out = (float*)d_out;

    // 1) support = 0.9*x + 0.1*h0
    {
        int total4 = (N * DFEAT) / 4;
        int blocks = (total4 + 255) / 256;
        gcnii_support_kernel<<<blocks, 256, 0, stream>>>(x, h0, sup, total4);
    }
    // 2) out = 0.5 * support @ W   (WMMA; also initializes every element of out)
    {
        int blocks = (N + 127) / 128;
        gcnii_gemm_kernel<<<blocks, 256, 0, stream>>>(sup, W, out, N);
    }
    // 3) out[row] += 0.5 * val * support[col]
    {
        long long threads = (long long)E * 16;
        int blocks = (int)((threads + 255) / 256);
        gcnii_edge_kernel<<<blocks, 256, 0, stream>>>(erow, ecol, eval, sup, out, E);
    }
}